// get_model_11330123727117
// MI455X (gfx1250) — compile-verified
//
#include <hip/hip_runtime.h>
#include <hip/hip_bf16.h>

// Problem constants (match reference)
#define NPTS    16384
#define NPOINT  1024
#define KNN     32
#define CH      256      // points per streamed chunk in kernel 2
#define NSAMP   10
#define NSEL    7
// SUBSET = round(32*0.9) = 29 -> each subset mask clears 3 of 32 bits

typedef __attribute__((ext_vector_type(2))) float v2f;
typedef __attribute__((ext_vector_type(8))) float v8f;

// ---------- wave32 primitives (gfx1250) ----------
__device__ __forceinline__ float bperm_f(float v, int srcLane) {
  int r = __builtin_amdgcn_ds_bpermute(srcLane << 2, __builtin_bit_cast(int, v));
  return __builtin_bit_cast(float, r);
}
__device__ __forceinline__ int bperm_i(int v, int srcLane) {
  return __builtin_amdgcn_ds_bpermute(srcLane << 2, v);
}
__device__ __forceinline__ float wave_sum_f(float v, int lane) {
  #pragma unroll
  for (int off = 16; off >= 1; off >>= 1) v += bperm_f(v, lane ^ off);
  return v;
}

// Subset masks: which of the 32 (distance-sorted) neighbors belong to
// perm_i[:29].  Placeholders for JAX fold_in(key(42), i) permutations;
// each clears exactly 3 bits (32 - SUBSET).
__device__ __constant__ unsigned kSubsetMask[NSAMP] = {
  0xFFFFFFFFu & ~((1u << 3) | (1u << 17) | (1u << 26)),
  0xFFFFFFFFu & ~((1u << 0) | (1u << 9)  | (1u << 31)),
  0xFFFFFFFFu & ~((1u << 5) | (1u << 12) | (1u << 22)),
  0xFFFFFFFFu & ~((1u << 7) | (1u << 14) | (1u << 30)),
  0xFFFFFFFFu & ~((1u << 1) | (1u << 19) | (1u << 27)),
  0xFFFFFFFFu & ~((1u << 4) | (1u << 11) | (1u << 25)),
  0xFFFFFFFFu & ~((1u << 2) | (1u << 16) | (1u << 29)),
  0xFFFFFFFFu & ~((1u << 8) | (1u << 20) | (1u << 24)),
  0xFFFFFFFFu & ~((1u << 6) | (1u << 13) | (1u << 28)),
  0xFFFFFFFFu & ~((1u << 10) | (1u << 18) | (1u << 23)),
};

// =====================================================================
// Kernel 1: Farthest-point sampling. 1 block per batch, 1024 threads,
// 16 points per thread held in registers. Winner's coordinates are
// carried through the reduction so no gather is needed.
// =====================================================================
__global__ __launch_bounds__(1024) void fps_kernel(const float* __restrict__ xyz,
                                                   float* __restrict__ centers) {
  const int b    = blockIdx.x;
  const int tid  = threadIdx.x;
  const int lane = tid & 31;
  const int wv   = tid >> 5;
  const float* X = xyz + (size_t)b * NPTS * 3;

  float px[16], py[16], pz[16], dist[16];
  #pragma unroll
  for (int i = 0; i < 16; ++i) {
    int n = tid + i * 1024;            // strided ownership
    px[i] = X[3 * n + 0];
    py[i] = X[3 * n + 1];
    pz[i] = X[3 * n + 2];
    dist[i] = 1e10f;
  }

  __shared__ float sc[3];              // current farthest point coords
  __shared__ float wD[32], wX[32], wY[32], wZ[32];
  __shared__ int   wI[32];
  if (tid == 0) { sc[0] = px[0]; sc[1] = py[0]; sc[2] = pz[0]; }  // farthest=0
  __syncthreads();

  float* C = centers + (size_t)b * NPOINT * 3;
  for (int it = 0; it < NPOINT; ++it) {
    float cx = sc[0], cy = sc[1], cz = sc[2];
    if (tid == 0) { C[3 * it + 0] = cx; C[3 * it + 1] = cy; C[3 * it + 2] = cz; }

    // local update + (max dist, min index) argmax
    float bD = -1.0f, bX = 0.f, bY = 0.f, bZ = 0.f;
    int bI = 0x7fffffff;
    #pragma unroll
    for (int i = 0; i < 16; ++i) {
      float dx = px[i] - cx, dy = py[i] - cy, dz = pz[i] - cz;
      float d = fminf(dist[i], dx * dx + dy * dy + dz * dz);
      dist[i] = d;
      int n = tid + i * 1024;
      if (d > bD || (d == bD && n < bI)) { bD = d; bI = n; bX = px[i]; bY = py[i]; bZ = pz[i]; }
    }
    // wave reduction
    #pragma unroll
    for (int off = 16; off >= 1; off >>= 1) {
      float oD = bperm_f(bD, lane ^ off);
      int   oI = bperm_i(bI, lane ^ off);
      float oX = bperm_f(bX, lane ^ off);
      float oY = bperm_f(bY, lane ^ off);
      float oZ = bperm_f(bZ, lane ^ off);
      if (oD > bD || (oD == bD && oI < bI)) { bD = oD; bI = oI; bX = oX; bY = oY; bZ = oZ; }
    }
    if (lane == 0) { wD[wv] = bD; wI[wv] = bI; wX[wv] = bX; wY[wv] = bY; wZ[wv] = bZ; }
    __syncthreads();
    if (wv == 0) {
      float d = wD[lane], x = wX[lane], y = wY[lane], z = wZ[lane];
      int i2 = wI[lane];
      #pragma unroll
      for (int off = 16; off >= 1; off >>= 1) {
        float oD = bperm_f(d, lane ^ off);
        int   oI = bperm_i(i2, lane ^ off);
        float oX = bperm_f(x, lane ^ off);
        float oY = bperm_f(y, lane ^ off);
        float oZ = bperm_f(z, lane ^ off);
        if (oD > d || (oD == d && oI < i2)) { d = oD; i2 = oI; x = oX; y = oY; z = oZ; }
      }
      if (lane == 0) { sc[0] = x; sc[1] = y; sc[2] = z; }
    }
    __syncthreads();
  }
}

// =====================================================================
// Kernel 2: fused KNN (WMMA f32 16x16x4 distance tiles) + streaming
// top-32 selection + robust centroid + 5 output features.
// Grid: B * (NPOINT/16) blocks, 512 threads (16 waves); wave w owns
// center row w of the 16-center tile.
// =====================================================================
__global__ __launch_bounds__(512) void knn_feat_kernel(const float* __restrict__ xyz,
                                                       const float* __restrict__ centers,
                                                       float* __restrict__ out) {
  const int b    = blockIdx.x >> 6;    // NPOINT/16 = 64 tiles per batch
  const int tile = blockIdx.x & 63;
  const int w    = threadIdx.x >> 5;
  const int lane = threadIdx.x & 31;
  const float* X = xyz + (size_t)b * NPTS * 3;

  __shared__ float cst[16][4];         // cx, cy, cz, |c|^2 for the 16 centers
  __shared__ float chD[16][CH];        // d^2 chunk buffer (16 rows x 256 cols)
  __shared__ float nb[16][KNN][3];     // sorted neighbor coords per wave
  __shared__ float clS[16][NSAMP][3];  // candidate centroids per wave
  __shared__ float pS[16][NSAMP];      // pairwise-distance sums

  if (threadIdx.x < 16) {
    const float* Cp = centers + ((size_t)b * NPOINT + tile * 16 + threadIdx.x) * 3;
    float x = Cp[0], y = Cp[1], z = Cp[2];
    cst[threadIdx.x][0] = x;
    cst[threadIdx.x][1] = y;
    cst[threadIdx.x][2] = z;
    cst[threadIdx.x][3] = x * x + y * y + z * z;
  }
  __syncthreads();

  // A fragment: 16x4 f32 (K=3 padded to 4). Lanes 0-15: K=0,1; 16-31: K=2,3.
  v2f afrag;
  if (lane < 16) { afrag.x = cst[lane][0];      afrag.y = cst[lane][1]; }
  else           { afrag.x = cst[lane - 16][2]; afrag.y = 0.0f; }
  // |c|^2 for the 8 rows this lane accumulates in the D fragment
  float crn[8];
  const int rbase = (lane < 16) ? 0 : 8;
  #pragma unroll
  for (int v = 0; v < 8; ++v) crn[v] = cst[rbase + v][3];

  // sorted top-32: lane L holds the L-th smallest (d2, idx) of its row
  float bestD = 3.4e38f;
  int   bestI = 0;

  for (int chunk = 0; chunk < NPTS / CH; ++chunk) {
    // ---- phase A: WMMA distance tile (wave w -> column tile w) ----
    int pidx = chunk * CH + w * 16 + (lane & 15);
    float pxv = X[3 * pidx + 0], pyv = X[3 * pidx + 1], pzv = X[3 * pidx + 2];
    float pn2 = pxv * pxv + pyv * pyv + pzv * pzv;
    v2f bfrag;
    if (lane < 16) { bfrag.x = pxv; bfrag.y = pyv; }
    else           { bfrag.x = pzv; bfrag.y = 0.0f; }
    v8f acc = {};
    acc = __builtin_amdgcn_wmma_f32_16x16x4_f32(
        /*neg_a=*/false, afrag, /*neg_b=*/false, bfrag,
        /*c_mod=*/(short)0, acc, /*reuse_a=*/false, /*reuse_b=*/false);
    int colBase = w * 16 + (lane & 15);
    #pragma unroll
    for (int v = 0; v < 8; ++v) {
      chD[rbase + v][colBase] = crn[v] + pn2 - 2.0f * acc[v];
    }
    __syncthreads();

    // ---- phase B: wave w merges its row's 256 candidates ----
    for (int g = 0; g < CH / 32; ++g) {
      float cand = chD[w][g * 32 + lane];
      int   base = chunk * CH + g * 32;
      float kth = bperm_f(bestD, 31);
      unsigned m = __builtin_amdgcn_ballot_w32(cand < kth);
      while (m) {
        int j = __builtin_ctz(m);
        m &= m - 1;
        float cd = bperm_f(cand, j);          // wave-uniform
        kth = bperm_f(bestD, 31);
        if (cd < kth) {                        // uniform branch
          unsigned le = __builtin_amdgcn_ballot_w32(bestD <= cd);
          int pos = (int)__builtin_popcount(le);
          int src = lane ? lane - 1 : 0;
          float uD = bperm_f(bestD, src);
          int   uI = bperm_i(bestI, src);
          if (lane > pos)       { bestD = uD; bestI = uI; }
          else if (lane == pos) { bestD = cd; bestI = base + j; }
        }
      }
    }
    __syncthreads();   // before chunk buffer is overwritten
  }

  // ---- robust centroid over the sorted 32 neighbors ----
  nb[w][lane][0] = X[3 * bestI + 0];
  nb[w][lane][1] = X[3 * bestI + 1];
  nb[w][lane][2] = X[3 * bestI + 2];
  __syncthreads();

  float clx = 0.f, cly = 0.f, clz = 0.f;
  if (lane < NSAMP) {
    unsigned msk = kSubsetMask[lane];
    #pragma unroll
    for (int j = 0; j < KNN; ++j) {
      if ((msk >> j) & 1u) { clx += nb[w][j][0]; cly += nb[w][j][1]; clz += nb[w][j][2]; }
    }
    const float inv29 = 1.0f / 29.0f;
    clx *= inv29; cly *= inv29; clz *= inv29;
    clS[w][lane][0] = clx; clS[w][lane][1] = cly; clS[w][lane][2] = clz;
  }
  __syncthreads();

  float psum = 0.f;
  if (lane < NSAMP) {
    for (int j = 0; j < NSAMP; ++j) {
      float dx = clx - clS[w][j][0];
      float dy = cly - clS[w][j][1];
      float dz = clz - clS[w][j][2];
      psum += dx * dx + dy * dy + dz * dz;
    }
    pS[w][lane] = psum;
  }
  __syncthreads();

  bool selme = false;
  if (lane < NSAMP) {
    int rank = 0;
    for (int j = 0; j < NSAMP; ++j) {
      float pj = pS[w][j];
      if (pj < psum || (pj == psum && j < lane)) ++rank;
    }
    selme = (rank < NSEL);
  }
  float ctx = wave_sum_f(selme ? clx : 0.f, lane) * (1.0f / NSEL);
  float cty = wave_sum_f(selme ? cly : 0.f, lane) * (1.0f / NSEL);
  float ctz = wave_sum_f(selme ? clz : 0.f, lane) * (1.0f / NSEL);

  // ---- 5 global features for center row w ----
  float nx = cst[w][0], ny = cst[w][1], nz = cst[w][2];
  float ref_norm = sqrtf(nx * nx + ny * ny + nz * nz);
  float inv = 1.0f / (ref_norm + 1e-4f);
  float ix = 0.2f * nx * inv + nx, iy = 0.2f * ny * inv + ny, iz = 0.2f * nz * inv + nz;
  float crx = nx - ctx, cry = ny - cty, crz = nz - ctz;
  float crd = sqrtf(crx * crx + cry * cry + crz * crz);
  float cvx = ix - ctx, cvy = iy - cty, cvz = iz - ctz;
  float cid = sqrtf(cvx * cvx + cvy * cvy + cvz * cvz);
  float ang_rci = (crx * cvx + cry * cvy + crz * cvz) / (crd * cid + 1e-6f);
  float irx = nx - ix, iry = ny - iy, irz = nz - iz;
  float icx = ctx - ix, icy = cty - iy, icz = ctz - iz;
  float ang_ric = (irx * icx + iry * icy + irz * icz) / (0.2f * cid + 1e-6f);

  if (lane == 0) {
    float* o = out + ((size_t)b * NPOINT + tile * 16 + w) * 5;
    o[0] = ref_norm; o[1] = crd; o[2] = cid; o[3] = ang_rci; o[4] = ang_ric;
  }
}

// =====================================================================
extern "C" void kernel_launch(void* const* d_in, const int* in_sizes, int n_in,
                              void* d_out, int out_size, void* d_ws, size_t ws_size,
                              hipStream_t stream) {
  const float* xyz = (const float*)d_in[0];     // (8, 16384, 3) f32
  float* centers   = (float*)d_ws;              // (8, 1024, 3) f32 scratch
  float* out       = (float*)d_out;             // (8, 1024, 5) f32

  fps_kernel<<<8, 1024, 0, stream>>>(xyz, centers);
  knn_feat_kernel<<<8 * (NPOINT / 16), 512, 0, stream>>>(xyz, centers, out);
}